// Criterion_64965675319882
// MI455X (gfx1250) — compile-verified
//
#include <hip/hip_runtime.h>
#include <hip/hip_bf16.h>
#include <math.h>

// ---------------------------------------------------------------------------
// Circle-loss + KD criterion for B=4096, D=128, f32.
//   d_in[0] = batch         [4096,128] f32 (row-normalized)
//   d_in[1] = teacher_batch [4096,128] f32
//   d_in[2] = labels        [4096] int32
//   d_in[3] = epoch         [1] int32
//   d_out   = {loss, loss_rank, loss_kd} f32
//   d_ws    : needs 2*B*B*4 bytes (sim + t_sim, 128 MB; fits MI455X 192MB L2)
//             + 3*B*4 bytes of per-row partials.
// ---------------------------------------------------------------------------

#define BSZ   4096
#define DDIM  128
#define POS_W 2.0f
#define NEG_W 40.0f
#define MARGIN 0.1f
#define THRESH 0.5f
#define TAU   4.0f
#define ALPHA 1.0f
#define BETA  0.5f
#define N_EPOCHS 100.0f

typedef __attribute__((ext_vector_type(2))) float v2f;
typedef __attribute__((ext_vector_type(8))) float v8f;

// ----------------------------------------------------------------------------
// Kernel 1: S = X * X^T via V_WMMA_F32_16X16X4_F32.
// One wave computes a 16x64 strip (4 column tiles) so the A fragment is loaded
// once per k-step and reused by 4 WMMAs: 1.25 loads/WMMA instead of 2.0.
// grid = (B/64, B/16, 2); block = 32 (one wave32). z=0: batch, z=1: teacher.
// ----------------------------------------------------------------------------
__global__ __launch_bounds__(32)
void gemm_xxt_wmma(const float* __restrict__ X0, const float* __restrict__ X1,
                   float* __restrict__ S0, float* __restrict__ S1)
{
    const float* __restrict__ X = (blockIdx.z == 0) ? X0 : X1;
    float* __restrict__ S       = (blockIdx.z == 0) ? S0 : S1;

    const int i0   = blockIdx.y << 4;   // 16-row strip
    const int j0   = blockIdx.x << 6;   // 64-col strip (4 tiles of 16)
    const int lane = threadIdx.x & 31;
    const int m    = lane & 15;         // row-in-tile (A) / col-in-tile (B)
    const int sel  = lane >> 4;         // lanes 16-31 hold K+2,K+3

    // A frag: lane holds A[i0+m][k + 2*sel .. +1]; B frag of X*X^T is rows of X
    // with the column tile: B[k][n] = X[jt+n][k] -> identical addressing.
    const float* __restrict__ arow = X + (size_t)(i0 + m) * DDIM + 2 * sel;
    const float* __restrict__ br0  = X + (size_t)(j0 +  0 + m) * DDIM + 2 * sel;
    const float* __restrict__ br1  = X + (size_t)(j0 + 16 + m) * DDIM + 2 * sel;
    const float* __restrict__ br2  = X + (size_t)(j0 + 32 + m) * DDIM + 2 * sel;
    const float* __restrict__ br3  = X + (size_t)(j0 + 48 + m) * DDIM + 2 * sel;

    v8f c0 = {}, c1 = {}, c2 = {}, c3 = {};
#pragma unroll
    for (int k = 0; k < DDIM; k += 4) {
        v2f a  = *(const v2f*)(arow + k);   // 8B aligned: k + 2*sel is even
        v2f b0 = *(const v2f*)(br0 + k);
        v2f b1 = *(const v2f*)(br1 + k);
        v2f b2 = *(const v2f*)(br2 + k);
        v2f b3 = *(const v2f*)(br3 + k);
        c0 = __builtin_amdgcn_wmma_f32_16x16x4_f32(false, a, false, b0, (short)0, c0, false, false);
        c1 = __builtin_amdgcn_wmma_f32_16x16x4_f32(false, a, false, b1, (short)0, c1, false, false);
        c2 = __builtin_amdgcn_wmma_f32_16x16x4_f32(false, a, false, b2, (short)0, c2, false, false);
        c3 = __builtin_amdgcn_wmma_f32_16x16x4_f32(false, a, false, b3, (short)0, c3, false, false);
    }

    // D layout: VGPR r -> row (r + 8*sel), col (lane&15)
    float* __restrict__ out = S + (size_t)(i0 + 8 * sel) * BSZ + (j0 + m);
#pragma unroll
    for (int r = 0; r < 8; ++r) {
        out[(size_t)r * BSZ +  0] = c0[r];
        out[(size_t)r * BSZ + 16] = c1[r];
        out[(size_t)r * BSZ + 32] = c2[r];
        out[(size_t)r * BSZ + 48] = c3[r];
    }
}

// ----------------------------------------------------------------------------
// Block-wide tree reduction (deterministic). blockDim.x == 256.
// op: 0 = sum, 1 = max, 2 = min
// ----------------------------------------------------------------------------
__device__ __forceinline__ float block_reduce(float v, float* red, int op)
{
    const int t = threadIdx.x;
    red[t] = v;
    __syncthreads();
#pragma unroll
    for (int s = 128; s > 0; s >>= 1) {
        if (t < s) {
            float a = red[t], b = red[t + s];
            red[t] = (op == 0) ? (a + b) : ((op == 1) ? fmaxf(a, b) : fminf(a, b));
        }
        __syncthreads();
    }
    float r = red[0];
    __syncthreads();
    return r;
}

// ----------------------------------------------------------------------------
// Kernel 2: per-row circle-loss terms + KD KL. One 256-thread block per row.
// Rows of sim / t_sim are staged in LDS once; three passes read from LDS.
// ----------------------------------------------------------------------------
__global__ __launch_bounds__(256)
void row_loss_kernel(const float* __restrict__ sim, const float* __restrict__ tsim,
                     const int* __restrict__ labels,
                     float* __restrict__ row_rank, float* __restrict__ row_valid,
                     float* __restrict__ row_kl)
{
    __shared__ float         s_sim[BSZ];
    __shared__ float         s_tsm[BSZ];
    __shared__ unsigned char s_neg[BSZ];
    __shared__ float         red[256];

    const int i   = blockIdx.x;
    const int tid = threadIdx.x;
    const int li  = labels[i];
    const float inv_tau = 1.0f / TAU;

    // ---- stage row i of both Gram matrices + label mask into LDS ----
    for (int j = tid; j < BSZ; j += 256) {
        s_sim[j] = sim[(size_t)i * BSZ + j];
        s_tsm[j] = tsim[(size_t)i * BSZ + j];
        s_neg[j] = (labels[j] != li) ? 1u : 0u;
    }
    __syncthreads();

    // ---- pass 1: mining extrema + softmax row maxima ----
    float minp =  INFINITY, maxn = -INFINITY;
    float mp   = -INFINITY, mq   = -INFINITY;
    for (int j = tid; j < BSZ; j += 256) {
        const float s   = s_sim[j];
        const bool  neg = s_neg[j] != 0;
        const bool  pos = !neg && (j != i);
        if (pos) minp = fminf(minp, s);
        if (neg) maxn = fmaxf(maxn, s);
        const float sc = neg ? BETA : 1.0f;
        mp = fmaxf(mp, s * sc * inv_tau);
        mq = fmaxf(mq, s_tsm[j] * sc * inv_tau);
    }
    minp = block_reduce(minp, red, 2);
    maxn = block_reduce(maxn, red, 1);
    mp   = block_reduce(mp,   red, 1);
    mq   = block_reduce(mq,   red, 1);

    // ---- pass 2: masked exp sums + softmax denominators ----
    float sep = 0.0f, sen = 0.0f, anyp = 0.0f, anyn = 0.0f;
    float sp  = 0.0f, sq  = 0.0f;
    for (int j = tid; j < BSZ; j += 256) {
        const float s   = s_sim[j];
        const bool  neg = s_neg[j] != 0;
        const bool  pos = !neg && (j != i);
        if (neg && (s + MARGIN > minp)) { sen += __expf(NEG_W * (s - THRESH)); anyn = 1.0f; }
        if (pos && (s - MARGIN < maxn)) { sep += __expf(-POS_W * (s - THRESH)); anyp = 1.0f; }
        const float sc = neg ? BETA : 1.0f;
        sp += __expf(s        * sc * inv_tau - mp);
        sq += __expf(s_tsm[j] * sc * inv_tau - mq);
    }
    sep  = block_reduce(sep,  red, 0);
    sen  = block_reduce(sen,  red, 0);
    anyp = block_reduce(anyp, red, 1);
    anyn = block_reduce(anyn, red, 1);
    sp   = block_reduce(sp,   red, 0);
    sq   = block_reduce(sq,   red, 0);

    // ---- pass 3: KL(q || p) over the row ----
    const float lsp = __logf(sp), lsq = __logf(sq);
    float kl = 0.0f;
    for (int j = tid; j < BSZ; j += 256) {
        const float sc   = (s_neg[j] != 0) ? BETA : 1.0f;
        const float xq   = s_tsm[j] * sc * inv_tau;
        const float logq = xq - mq - lsq;
        const float logp = s_sim[j] * sc * inv_tau - mp - lsp;
        kl += __expf(logq) * (logq - logp);
    }
    kl = block_reduce(kl, red, 0);

    if (tid == 0) {
        const bool valid = (anyn > 0.0f) && (anyp > 0.0f);
        const float pos_term = (1.0f / POS_W) * log1pf(sep);
        const float neg_term = (1.0f / NEG_W) * log1pf(sen);
        row_rank[i]  = valid ? (pos_term + neg_term) : 0.0f;
        row_valid[i] = valid ? 1.0f : 0.0f;
        row_kl[i]    = kl;
    }
}

// ----------------------------------------------------------------------------
// Kernel 3: fold per-row partials into the three scalar outputs. One block.
// ----------------------------------------------------------------------------
__global__ __launch_bounds__(256)
void finalize_kernel(const float* __restrict__ row_rank, const float* __restrict__ row_valid,
                     const float* __restrict__ row_kl, const int* __restrict__ epoch,
                     float* __restrict__ out)
{
    __shared__ float red[256];
    const int tid = threadIdx.x;
    float sr = 0.0f, sv = 0.0f, sk = 0.0f;
    for (int r = tid; r < BSZ; r += 256) {
        sr += row_rank[r];
        sv += row_valid[r];
        sk += row_kl[r];
    }
    sr = block_reduce(sr, red, 0);
    sv = block_reduce(sv, red, 0);
    sk = block_reduce(sk, red, 0);
    if (tid == 0) {
        const float n_valid   = fmaxf(sv, 1.0f);
        const float loss_rank = sr / n_valid;
        const float loss_kd   = sk / (float)BSZ;
        const float w         = ((float)epoch[0] / N_EPOCHS) * ALPHA * (TAU * TAU);
        out[0] = loss_rank + w * loss_kd;
        out[1] = loss_rank;
        out[2] = loss_kd;
    }
}

// ----------------------------------------------------------------------------
extern "C" void kernel_launch(void* const* d_in, const int* in_sizes, int n_in,
                              void* d_out, int out_size, void* d_ws, size_t ws_size,
                              hipStream_t stream)
{
    const float* batch   = (const float*)d_in[0];
    const float* teacher = (const float*)d_in[1];
    const int*   labels  = (const int*)d_in[2];
    const int*   epoch   = (const int*)d_in[3];
    float*       out     = (float*)d_out;

    float* sim       = (float*)d_ws;                 // [B,B]
    float* tsim      = sim  + (size_t)BSZ * BSZ;     // [B,B]
    float* row_rank  = tsim + (size_t)BSZ * BSZ;     // [B]
    float* row_valid = row_rank  + BSZ;              // [B]
    float* row_kl    = row_valid + BSZ;              // [B]

    // 1) both Gram matrices via f32 WMMA (z=0: student, z=1: teacher)
    dim3 ggrid(BSZ / 64, BSZ / 16, 2);
    gemm_xxt_wmma<<<ggrid, 32, 0, stream>>>(batch, teacher, sim, tsim);

    // 2) per-row mining / LSE / KL terms (rows staged in LDS)
    row_loss_kernel<<<BSZ, 256, 0, stream>>>(sim, tsim, labels,
                                             row_rank, row_valid, row_kl);

    // 3) scalar outputs
    finalize_kernel<<<1, 256, 0, stream>>>(row_rank, row_valid, row_kl, epoch, out);
}